// LightGCN_11115375362611
// MI455X (gfx1250) — compile-verified
//
#include <hip/hip_runtime.h>
#include <hip/hip_bf16.h>

// ---------------- problem constants (match reference) ----------------
#define N_USERS   400000
#define N_ITEMS   200000
#define DIM       64
#define N_NODES   (N_USERS + N_ITEMS)          // 600000
#define NNZ       1200000                      // 2*E
#define BATCH     8192
#define R4        (DIM / 4)                    // 16 float4 per row
#define NODE_F4   ((size_t)N_NODES * R4)       // 9,600,000 float4
#define USER_F4   ((size_t)N_USERS * R4)       // 6,400,000 float4

// CDNA5 async global->LDS path (builtin confirmed present, 4 args:
// (global int*, lds int*, imm offset, imm cpol)).
#if defined(__gfx1250__) && __has_builtin(__builtin_amdgcn_global_load_async_to_lds_b32) && __has_builtin(__builtin_amdgcn_s_wait_asynccnt)
#define HAVE_ASYNC_LDS 1
#else
#define HAVE_ASYNC_LDS 0
#endif

typedef __attribute__((address_space(1))) int   g1i_t;
typedef __attribute__((address_space(3))) int   l3i_t;

__device__ __forceinline__ float4 mask_sel(float4 e, int4 m) {
    float4 v;
    v.x = m.x ? e.x : 0.0f;
    v.y = m.y ? e.y : 0.0f;
    v.z = m.z ? e.z : 0.0f;
    v.w = m.w ? e.w : 0.0f;
    return v;
}

// ---------------- init (4-buffer scheme): x0 = masked concat; y1..y3 = 0 ----
__global__ __launch_bounds__(256)
void k_init4(const float4* __restrict__ ue4, const float4* __restrict__ ie4,
             const int4* __restrict__ um4, const int4* __restrict__ im4,
             float4* __restrict__ x0, float4* __restrict__ y1,
             float4* __restrict__ y2, float4* __restrict__ y3) {
    size_t i = (size_t)blockIdx.x * blockDim.x + threadIdx.x;
    if (i >= NODE_F4) return;
    float4 e; int4 m;
    if (i < USER_F4) { e = ue4[i];            m = um4[i]; }
    else             { e = ie4[i - USER_F4];  m = im4[i - USER_F4]; }
    float4 z = make_float4(0.f, 0.f, 0.f, 0.f);
    x0[i] = mask_sel(e, m);
    y1[i] = z;
    y2[i] = z;
    y3[i] = z;
}

// ---------------- init (3-buffer fallback): x = acc = masked; y = 0 ---------
__global__ __launch_bounds__(256)
void k_init3(const float4* __restrict__ ue4, const float4* __restrict__ ie4,
             const int4* __restrict__ um4, const int4* __restrict__ im4,
             float4* __restrict__ x, float4* __restrict__ acc, float4* __restrict__ y) {
    size_t i = (size_t)blockIdx.x * blockDim.x + threadIdx.x;
    if (i >= NODE_F4) return;
    float4 e; int4 m;
    if (i < USER_F4) { e = ue4[i];            m = um4[i]; }
    else             { e = ie4[i - USER_F4];  m = im4[i - USER_F4]; }
    float4 v = mask_sel(e, m);
    x[i]   = v;
    acc[i] = v;
    y[i]   = make_float4(0.f, 0.f, 0.f, 0.f);
}

// ---------------- COO SpMM, edges staged in LDS via async copy --------------
// Block = 256 threads (8 wave32). Block owns EPB edges; each wave processes
// EPB/8 edges; lane l owns floats [2l, 2l+1] of the 64-wide row.
#define EPB 64
__global__ __launch_bounds__(256)
void k_spmm(const int* __restrict__ adj_row, const int* __restrict__ adj_col,
            const float* __restrict__ adj_val,
            const float* __restrict__ xin, float* __restrict__ yout) {
    __shared__ int   sRow[EPB];
    __shared__ int   sCol[EPB];
    __shared__ float sVal[EPB];

    const size_t base = (size_t)blockIdx.x * EPB;
    const int t = threadIdx.x;

    if (t < EPB) {
        size_t e = base + t;
#if HAVE_ASYNC_LDS
        // async data-mover: stream edge triples into LDS, tracked by ASYNCcnt
        // (decoupled from the LOADcnt gathers below). b32 copy is type-
        // agnostic, so the float array goes through int pointers.
        __builtin_amdgcn_global_load_async_to_lds_b32((g1i_t*)(adj_row + e), (l3i_t*)(sRow + t), 0, 0);
        __builtin_amdgcn_global_load_async_to_lds_b32((g1i_t*)(adj_col + e), (l3i_t*)(sCol + t), 0, 0);
        __builtin_amdgcn_global_load_async_to_lds_b32((g1i_t*)(adj_val + e), (l3i_t*)(sVal + t), 0, 0);
#else
        sRow[t] = adj_row[e];
        sCol[t] = adj_col[e];
        sVal[t] = adj_val[e];
#endif
    }
#if HAVE_ASYNC_LDS
    __builtin_amdgcn_s_wait_asynccnt(0);
#endif
    __syncthreads();

    const int wave = t >> 5;
    const int lane = t & 31;
#pragma unroll
    for (int k = 0; k < EPB / 8; ++k) {
        const int li  = wave * (EPB / 8) + k;
        const int row = sRow[li];
        const int col = sCol[li];
        const float val = sVal[li];
        // one coalesced 256B gather per edge per wave (float2 per lane)
        const float2 xv = ((const float2*)(xin + (size_t)col * DIM))[lane];
        float* yp = yout + (size_t)row * DIM + 2 * lane;
        atomicAdd(yp,     val * xv.x);   // global_atomic_add_f32, resolves in L2
        atomicAdd(yp + 1, val * xv.y);
    }
}

// ---------------- 3-buffer fallback: acc += y; optionally zero next buffer --
__global__ __launch_bounds__(256)
void k_accum(const float4* __restrict__ y, float4* __restrict__ acc,
             float4* __restrict__ zbuf, int do_zero) {
    size_t i = (size_t)blockIdx.x * blockDim.x + threadIdx.x;
    if (i >= NODE_F4) return;
    float4 a = acc[i];
    float4 b = y[i];
    a.x += b.x; a.y += b.y; a.z += b.z; a.w += b.w;
    acc[i] = a;
    if (do_zero) zbuf[i] = make_float4(0.f, 0.f, 0.f, 0.f);
}

// ---------------- gather (4-buffer): 0.25*(x0+y1+y2+y3) rows + masked ego ---
// out layout: [users_emb | pos_emb | neg_emb | users_ego | pos_ego | neg_ego]
__global__ __launch_bounds__(256)
void k_gather4(const float4* __restrict__ x0, const float4* __restrict__ y1,
               const float4* __restrict__ y2, const float4* __restrict__ y3,
               const float4* __restrict__ ue4, const float4* __restrict__ ie4,
               const int4* __restrict__ um4, const int4* __restrict__ im4,
               const int* __restrict__ users, const int* __restrict__ pos,
               const int* __restrict__ neg, float4* __restrict__ out4) {
    int idx = blockIdx.x * blockDim.x + threadIdx.x;
    const int SEC = BATCH * R4;
    if (idx >= 6 * SEC) return;
    int sec = idx / SEC;
    int rem = idx - sec * SEC;
    int b = rem / R4;
    int c = rem - b * R4;
    float4 v;
    if (sec < 3) {
        int node = (sec == 0) ? users[b]
                 : (sec == 1) ? (N_USERS + pos[b])
                              : (N_USERS + neg[b]);
        size_t o = (size_t)node * R4 + c;
        float4 a0 = x0[o], a1 = y1[o], a2 = y2[o], a3 = y3[o];
        v = make_float4(0.25f * (a0.x + a1.x + a2.x + a3.x),
                        0.25f * (a0.y + a1.y + a2.y + a3.y),
                        0.25f * (a0.z + a1.z + a2.z + a3.z),
                        0.25f * (a0.w + a1.w + a2.w + a3.w));
    } else if (sec == 3) {
        int u = users[b];
        v = mask_sel(ue4[(size_t)u * R4 + c], um4[(size_t)u * R4 + c]);
    } else {
        int it = (sec == 4) ? pos[b] : neg[b];
        v = mask_sel(ie4[(size_t)it * R4 + c], im4[(size_t)it * R4 + c]);
    }
    out4[idx] = v;
}

// ---------------- gather (3-buffer): 0.25*acc rows + masked ego -------------
__global__ __launch_bounds__(256)
void k_gather3(const float4* __restrict__ acc4,
               const float4* __restrict__ ue4, const float4* __restrict__ ie4,
               const int4* __restrict__ um4, const int4* __restrict__ im4,
               const int* __restrict__ users, const int* __restrict__ pos,
               const int* __restrict__ neg, float4* __restrict__ out4) {
    int idx = blockIdx.x * blockDim.x + threadIdx.x;
    const int SEC = BATCH * R4;
    if (idx >= 6 * SEC) return;
    int sec = idx / SEC;
    int rem = idx - sec * SEC;
    int b = rem / R4;
    int c = rem - b * R4;
    float4 v;
    if (sec < 3) {
        int node = (sec == 0) ? users[b]
                 : (sec == 1) ? (N_USERS + pos[b])
                              : (N_USERS + neg[b]);
        float4 a = acc4[(size_t)node * R4 + c];
        v = make_float4(0.25f * a.x, 0.25f * a.y, 0.25f * a.z, 0.25f * a.w);
    } else if (sec == 3) {
        int u = users[b];
        v = mask_sel(ue4[(size_t)u * R4 + c], um4[(size_t)u * R4 + c]);
    } else {
        int it = (sec == 4) ? pos[b] : neg[b];
        v = mask_sel(ie4[(size_t)it * R4 + c], im4[(size_t)it * R4 + c]);
    }
    out4[idx] = v;
}

// ---------------- host launcher ----------------
extern "C" void kernel_launch(void* const* d_in, const int* in_sizes, int n_in,
                              void* d_out, int out_size, void* d_ws, size_t ws_size,
                              hipStream_t stream) {
    const float* user_emb  = (const float*)d_in[0];
    const float* item_emb  = (const float*)d_in[1];
    const int*   user_mask = (const int*)  d_in[2];
    const int*   item_mask = (const int*)  d_in[3];
    const int*   adj_row   = (const int*)  d_in[4];
    const int*   adj_col   = (const int*)  d_in[5];
    const float* adj_val   = (const float*)d_in[6];
    const int*   users     = (const int*)  d_in[7];
    const int*   pos_items = (const int*)  d_in[8];
    const int*   neg_items = (const int*)  d_in[9];
    float* out = (float*)d_out;

    const size_t NF = (size_t)N_NODES * DIM;     // 38,400,000 floats / buffer
    float* ws = (float*)d_ws;

    const int initBlocks = (int)((NODE_F4 + 255) / 256);   // 37500
    const int spmmBlocks = NNZ / EPB;                      // 18750
    const int gathBlocks = (6 * BATCH * R4 + 255) / 256;   // 3072

    if (ws_size >= 4 * NF * sizeof(float)) {
        // -------- fast scheme: keep all layer outputs, no accumulation pass --
        float* x0 = ws;
        float* y1 = ws + NF;
        float* y2 = ws + 2 * NF;
        float* y3 = ws + 3 * NF;
        k_init4<<<initBlocks, 256, 0, stream>>>((const float4*)user_emb, (const float4*)item_emb,
                                                (const int4*)user_mask, (const int4*)item_mask,
                                                (float4*)x0, (float4*)y1, (float4*)y2, (float4*)y3);
        k_spmm<<<spmmBlocks, 256, 0, stream>>>(adj_row, adj_col, adj_val, x0, y1);
        k_spmm<<<spmmBlocks, 256, 0, stream>>>(adj_row, adj_col, adj_val, y1, y2);
        k_spmm<<<spmmBlocks, 256, 0, stream>>>(adj_row, adj_col, adj_val, y2, y3);
        k_gather4<<<gathBlocks, 256, 0, stream>>>((const float4*)x0, (const float4*)y1,
                                                  (const float4*)y2, (const float4*)y3,
                                                  (const float4*)user_emb, (const float4*)item_emb,
                                                  (const int4*)user_mask, (const int4*)item_mask,
                                                  users, pos_items, neg_items, (float4*)out);
    } else {
        // -------- 3-buffer fallback: ping-pong + running accumulator ---------
        float* bufA = ws;
        float* bufB = ws + NF;
        float* acc  = ws + 2 * NF;
        k_init3<<<initBlocks, 256, 0, stream>>>((const float4*)user_emb, (const float4*)item_emb,
                                                (const int4*)user_mask, (const int4*)item_mask,
                                                (float4*)bufA, (float4*)acc, (float4*)bufB);
        k_spmm<<<spmmBlocks, 256, 0, stream>>>(adj_row, adj_col, adj_val, bufA, bufB);
        k_accum<<<initBlocks, 256, 0, stream>>>((const float4*)bufB, (float4*)acc, (float4*)bufA, 1);
        k_spmm<<<spmmBlocks, 256, 0, stream>>>(adj_row, adj_col, adj_val, bufB, bufA);
        k_accum<<<initBlocks, 256, 0, stream>>>((const float4*)bufA, (float4*)acc, (float4*)bufB, 1);
        k_spmm<<<spmmBlocks, 256, 0, stream>>>(adj_row, adj_col, adj_val, bufA, bufB);
        k_accum<<<initBlocks, 256, 0, stream>>>((const float4*)bufB, (float4*)acc, (float4*)bufB, 0);
        k_gather3<<<gathBlocks, 256, 0, stream>>>((const float4*)acc,
                                                  (const float4*)user_emb, (const float4*)item_emb,
                                                  (const int4*)user_mask, (const int4*)item_mask,
                                                  users, pos_items, neg_items, (float4*)out);
    }
}